// StationFlowGCN_63015760166986
// MI455X (gfx1250) — compile-verified
//
#include <hip/hip_runtime.h>

// ---------------------------------------------------------------------------
// CDNA5 (gfx1250) fp32 WMMA helpers
// ---------------------------------------------------------------------------
typedef float v2f __attribute__((ext_vector_type(2)));
typedef float v8f __attribute__((ext_vector_type(8)));

__device__ __forceinline__ v8f wmma4(v2f a, v2f b, v8f c) {
    // D = A(16x4 f32) * B(4x16 f32) + C(16x16 f32) -- exact fp32 matrix op
    return __builtin_amdgcn_wmma_f32_16x16x4_f32(
        false, a, false, b, (short)0, c, false, false);
}

__device__ __forceinline__ v8f zero8() {
    v8f z = {0.f, 0.f, 0.f, 0.f, 0.f, 0.f, 0.f, 0.f};
    return z;
}

__device__ __forceinline__ v2f ld2g(const float* p) {   // 8B global load
    v2f r;
    const float2 t = *reinterpret_cast<const float2*>(p);
    r.x = t.x; r.y = t.y;
    return r;
}

__device__ __forceinline__ v2f ld2s(const float* p) {   // 8B LDS load
    v2f r;
    const float2 t = *reinterpret_cast<const float2*>(p);
    r.x = t.x; r.y = t.y;
    return r;
}

// Per-wave LDS producer->consumer ordering (DS ops are in-order per wave on
// CDNA5; this stops the compiler from reordering across the phase boundary
// and drains DScnt explicitly).
__device__ __forceinline__ void wave_lds_fence() {
    __builtin_amdgcn_wave_barrier();
    asm volatile("s_wait_dscnt 0" ::: "memory");
    __builtin_amdgcn_wave_barrier();
}

__device__ __forceinline__ void gatomic_add(float* p, float v) {
    unsafeAtomicAdd(p, v);   // global_atomic_add_f32
}

// ---------------------------------------------------------------------------
// Weight pair-packing: Wp[2*(p*N+n)+{0,1}] = {W[2p][n], W[2p+1][n]}
// Makes every WMMA B-fragment a contiguous 8-byte load.
// ---------------------------------------------------------------------------
__global__ void k_pack_pairs(const float* __restrict__ W,
                             float* __restrict__ Wp, int Kdim, int Ndim) {
    int idx = blockIdx.x * blockDim.x + threadIdx.x;
    const int total = (Kdim >> 1) * Ndim;
    if (idx >= total) return;
    const int p = idx / Ndim;
    const int n = idx - p * Ndim;
    Wp[2 * idx + 0] = W[(2 * p) * Ndim + n];
    Wp[2 * idx + 1] = W[(2 * p + 1) * Ndim + n];
}

// ---------------------------------------------------------------------------
// Degree / normalization kernels (gcn_norm with self loops)
// ---------------------------------------------------------------------------
__global__ void k_deg_init(float* __restrict__ deg, int n) {
    int i = blockIdx.x * blockDim.x + threadIdx.x;
    if (i < n) deg[i] = 1.0f;                       // self-loop weight
}

__global__ void k_deg_acc(const long long* __restrict__ dst,
                          const float* __restrict__ ew,
                          float* __restrict__ deg, int E) {
    int e = blockIdx.x * blockDim.x + threadIdx.x;
    if (e < E) gatomic_add(&deg[(size_t)dst[e]], ew[e]);
}

__global__ void k_rsqrt_inplace(float* __restrict__ deg, int n) {
    int i = blockIdx.x * blockDim.x + threadIdx.x;
    if (i < n) {
        float d = deg[i];
        deg[i] = (d > 0.f) ? rsqrtf(d) : 0.f;
    }
}

__global__ void k_norm(const long long* __restrict__ src,
                       const long long* __restrict__ dst,
                       const float* __restrict__ ew,
                       const float* __restrict__ dinv,
                       float* __restrict__ norm, int E) {
    int e = blockIdx.x * blockDim.x + threadIdx.x;
    if (e < E) norm[e] = dinv[(size_t)src[e]] * ew[e] * dinv[(size_t)dst[e]];
}

// ---------------------------------------------------------------------------
// GEMM: C[M,N] = A[M,K] @ W[K,N] via V_WMMA_F32_16X16X4_F32.
// One wave computes one 16-row stripe of C across all N. KT = K/4, NT = N/16.
// A-frag: lane l holds row (l&15), K pair at k + 2*(l>>4) (contiguous float2).
// B-frag: lane l holds col (l&15); pair comes from packed Wp (one b64 load).
// D layout: VGPR v, lane l -> row v + 8*(l>>4), col (l&15).
// ---------------------------------------------------------------------------
template <int KT, int NT>
__global__ __launch_bounds__(256) void k_gemm(const float* __restrict__ A,
                                              const float* __restrict__ Wp,
                                              float* __restrict__ C, int M) {
    constexpr int Kdim = KT * 4;
    constexpr int Ndim = NT * 16;
    const int lane = threadIdx.x & 31;
    const int wave = threadIdx.x >> 5;
    const int col  = lane & 15;            // also A row within tile
    const int kh   = (lane >> 4) << 1;     // K sub-offset: 0 or 2
    const int khp  = lane >> 4;            // K pair sub-index: 0 or 1
    const int rb   = (lane >> 4) << 3;     // D row base: 0 or 8
    const int mt   = blockIdx.x * 8 + wave;
    const int mtiles = (M + 15) >> 4;
    if (mt >= mtiles) return;

    int m = mt * 16 + col;
    if (m >= M) m = M - 1;                 // clamp (loads only)

    v2f af[KT];
#pragma unroll
    for (int kt = 0; kt < KT; ++kt)
        af[kt] = ld2g(A + (size_t)m * Kdim + kt * 4 + kh);

#pragma unroll
    for (int nt = 0; nt < NT; ++nt) {
        v8f acc = zero8();
#pragma unroll
        for (int kt = 0; kt < KT; ++kt) {
            const size_t pidx = (size_t)(kt * 2 + khp) * Ndim + nt * 16 + col;
            acc = wmma4(af[kt], ld2g(Wp + 2 * pidx), acc);
        }
#pragma unroll
        for (int v = 0; v < 8; ++v) {
            const int r = mt * 16 + rb + v;
            if (r < M) C[(size_t)r * Ndim + nt * 16 + col] = acc[v];
        }
    }
}

// ---------------------------------------------------------------------------
// GCN aggregation (float4-vectorized).  dout/4 == 1<<qlog.
// ---------------------------------------------------------------------------
__global__ void k_selfinit(const float* __restrict__ t,
                           const float* __restrict__ dinv,
                           float* __restrict__ agg, int nq, int qlog) {
    int idx = blockIdx.x * blockDim.x + threadIdx.x;
    if (idx >= nq) return;
    const int i = idx >> qlog;                // node
    const float dv = dinv[i];
    const float s = dv * dv;                  // self-loop norm
    const float4 tv = *reinterpret_cast<const float4*>(t + (size_t)idx * 4);
    float4 o;
    o.x = tv.x * s; o.y = tv.y * s; o.z = tv.z * s; o.w = tv.w * s;
    *reinterpret_cast<float4*>(agg + (size_t)idx * 4) = o;
}

__global__ void k_scatter(const float* __restrict__ t,
                          const long long* __restrict__ src,
                          const long long* __restrict__ dst,
                          const float* __restrict__ norm,
                          float* __restrict__ agg,
                          int nq, int qlog, int dout) {
    int idx = blockIdx.x * blockDim.x + threadIdx.x;
    if (idx >= nq) return;
    const int e = idx >> qlog;
    const int q = (idx & ((1 << qlog) - 1)) << 2;
    const float nw = norm[e];
    const float4 tv =
        *reinterpret_cast<const float4*>(t + (size_t)src[e] * dout + q);
    float* ap = agg + (size_t)dst[e] * dout + q;
    gatomic_add(ap + 0, tv.x * nw);
    gatomic_add(ap + 1, tv.y * nw);
    gatomic_add(ap + 2, tv.z * nw);
    gatomic_add(ap + 3, tv.w * nw);
}

__global__ void k_bias_relu(float* __restrict__ agg,
                            const float* __restrict__ b, int nq, int qlog) {
    int idx = blockIdx.x * blockDim.x + threadIdx.x;
    if (idx >= nq) return;
    const int c = (idx & ((1 << qlog) - 1)) << 2;
    const float4 bv = *reinterpret_cast<const float4*>(b + c);
    float4 a = *reinterpret_cast<float4*>(agg + (size_t)idx * 4);
    a.x = fmaxf(a.x + bv.x, 0.f);
    a.y = fmaxf(a.y + bv.y, 0.f);
    a.z = fmaxf(a.z + bv.z, 0.f);
    a.w = fmaxf(a.w + bv.w, 0.f);
    *reinterpret_cast<float4*>(agg + (size_t)idx * 4) = a;
}

// ---------------------------------------------------------------------------
// Fused edge head: per 16-edge tile per wave:
//   A0 = concat(h[src], h[dst]) [16x64] (gather fused into A-frags)
//   relu(A0@W1+b1) [16x64] -> LDS tile -> relu(@W2+b2) [16x32]
//   -> relu(@W3+b3) [16x32] -> relu(@W4+b4) [16x12] -> global out.
// Weights staged in LDS in pair-interleaved (float2) layout so every
// B-fragment is a single ds_load_b64 into an adjacent VGPR pair.
// ---------------------------------------------------------------------------
__global__ __launch_bounds__(256) void k_edge_head(
    const float* __restrict__ h, const long long* __restrict__ src,
    const long long* __restrict__ dst,
    const float* __restrict__ W1, const float* __restrict__ b1,
    const float* __restrict__ W2, const float* __restrict__ b2,
    const float* __restrict__ W3, const float* __restrict__ b3,
    const float* __restrict__ W4, const float* __restrict__ b4,
    float* __restrict__ out, int E) {
    __shared__ float2 sW1p[32 * 64];          // 64x64  -> 16KB
    __shared__ float2 sW2p[32 * 32];          // 64x32  ->  8KB
    __shared__ float2 sW3p[16 * 32];          // 32x32  ->  4KB
    __shared__ float2 sW4p[16 * 16];          // 32x16p ->  2KB
    __shared__ float sB[144];                 // b1[64] b2[32] b3[32] b4pad[16]
    __shared__ __align__(16) float sTile[8][16 * 64];   // 32KB

    for (int i = threadIdx.x; i < 32 * 64; i += 256) {
        const int p = i >> 6, n = i & 63;
        float2 w;
        w.x = W1[(2 * p) * 64 + n];
        w.y = W1[(2 * p + 1) * 64 + n];
        sW1p[i] = w;
    }
    for (int i = threadIdx.x; i < 32 * 32; i += 256) {
        const int p = i >> 5, n = i & 31;
        float2 w;
        w.x = W2[(2 * p) * 32 + n];
        w.y = W2[(2 * p + 1) * 32 + n];
        sW2p[i] = w;
    }
    for (int i = threadIdx.x; i < 16 * 32; i += 256) {
        const int p = i >> 5, n = i & 31;
        float2 w;
        w.x = W3[(2 * p) * 32 + n];
        w.y = W3[(2 * p + 1) * 32 + n];
        sW3p[i] = w;
    }
    for (int i = threadIdx.x; i < 16 * 16; i += 256) {
        const int p = i >> 4, n = i & 15;
        float2 w;
        w.x = (n < 12) ? W4[(2 * p) * 12 + n] : 0.f;
        w.y = (n < 12) ? W4[(2 * p + 1) * 12 + n] : 0.f;
        sW4p[i] = w;
    }
    if (threadIdx.x < 144) {
        int i = threadIdx.x;
        float v;
        if (i < 64)       v = b1[i];
        else if (i < 96)  v = b2[i - 64];
        else if (i < 128) v = b3[i - 96];
        else              v = ((i - 128) < 12) ? b4[i - 128] : 0.f;
        sB[i] = v;
    }
    __syncthreads();

    const int lane = threadIdx.x & 31;
    const int wave = threadIdx.x >> 5;
    const int col  = lane & 15;
    const int kh   = (lane >> 4) << 1;
    const int khp  = lane >> 4;
    const int rb   = (lane >> 4) << 3;
    const int tile = blockIdx.x * 8 + wave;
    const int etiles = (E + 15) >> 4;
    if (tile >= etiles) return;

    int erow = tile * 16 + col;
    if (erow >= E) erow = E - 1;             // clamp for loads
    const size_t sroff = (size_t)src[erow] * 32;
    const size_t droff = (size_t)dst[erow] * 32;
    float* tp = &sTile[wave][0];

    // ---- stage 1: [16x64] = concat @ W1(64x64), relu ----
    v8f acc[4];
#pragma unroll
    for (int nt = 0; nt < 4; ++nt) acc[nt] = zero8();
#pragma unroll
    for (int kt = 0; kt < 16; ++kt) {
        const int k = kt * 4 + kh;
        const float* ap = (k < 32) ? (h + sroff + k) : (h + droff + (k - 32));
        const v2f a = ld2g(ap);
        const int pr = kt * 2 + khp;
#pragma unroll
        for (int nt = 0; nt < 4; ++nt)
            acc[nt] = wmma4(
                a, ld2s((const float*)&sW1p[pr * 64 + nt * 16 + col]),
                acc[nt]);
    }
#pragma unroll
    for (int nt = 0; nt < 4; ++nt) {
        const float bb = sB[nt * 16 + col];
#pragma unroll
        for (int v = 0; v < 8; ++v)
            tp[(rb + v) * 64 + nt * 16 + col] = fmaxf(acc[nt][v] + bb, 0.f);
    }
    wave_lds_fence();

    // ---- stage 2: [16x32] = tile(16x64) @ W2(64x32), relu ----
    v8f acc2[2] = {zero8(), zero8()};
#pragma unroll
    for (int kt = 0; kt < 16; ++kt) {
        const int k = kt * 4 + kh;
        const v2f a = ld2s(tp + col * 64 + k);
        const int pr = kt * 2 + khp;
#pragma unroll
        for (int nt = 0; nt < 2; ++nt)
            acc2[nt] = wmma4(
                a, ld2s((const float*)&sW2p[pr * 32 + nt * 16 + col]),
                acc2[nt]);
    }
    wave_lds_fence();
#pragma unroll
    for (int nt = 0; nt < 2; ++nt) {
        const float bb = sB[64 + nt * 16 + col];
#pragma unroll
        for (int v = 0; v < 8; ++v)
            tp[(rb + v) * 64 + nt * 16 + col] = fmaxf(acc2[nt][v] + bb, 0.f);
    }
    wave_lds_fence();

    // ---- stage 3: [16x32] = tile(16x32) @ W3(32x32), relu ----
    v8f acc3[2] = {zero8(), zero8()};
#pragma unroll
    for (int kt = 0; kt < 8; ++kt) {
        const int k = kt * 4 + kh;
        const v2f a = ld2s(tp + col * 64 + k);
        const int pr = kt * 2 + khp;
#pragma unroll
        for (int nt = 0; nt < 2; ++nt)
            acc3[nt] = wmma4(
                a, ld2s((const float*)&sW3p[pr * 32 + nt * 16 + col]),
                acc3[nt]);
    }
    wave_lds_fence();
#pragma unroll
    for (int nt = 0; nt < 2; ++nt) {
        const float bb = sB[96 + nt * 16 + col];
#pragma unroll
        for (int v = 0; v < 8; ++v)
            tp[(rb + v) * 64 + nt * 16 + col] = fmaxf(acc3[nt][v] + bb, 0.f);
    }
    wave_lds_fence();

    // ---- stage 4: [16x16(12)] = tile(16x32) @ W4(32x16pad), relu, store ----
    v8f acc4 = zero8();
#pragma unroll
    for (int kt = 0; kt < 8; ++kt) {
        const int k = kt * 4 + kh;
        const v2f a = ld2s(tp + col * 64 + k);
        const int pr = kt * 2 + khp;
        acc4 = wmma4(a, ld2s((const float*)&sW4p[pr * 16 + col]), acc4);
    }
    const float bb = sB[128 + col];
#pragma unroll
    for (int v = 0; v < 8; ++v) {
        const int r = tile * 16 + rb + v;
        if (r < E && col < 12)
            out[(size_t)r * 12 + col] = fmaxf(acc4[v] + bb, 0.f);
    }
}

// ---------------------------------------------------------------------------
// Fused node head: relu(relu(h@W1+b1)@W2+b2), h:[N,32], out:[N,12]
// ---------------------------------------------------------------------------
__global__ __launch_bounds__(256) void k_node_head(
    const float* __restrict__ h,
    const float* __restrict__ W1, const float* __restrict__ b1,
    const float* __restrict__ W2, const float* __restrict__ b2,
    float* __restrict__ out, int M) {
    __shared__ float2 sW1p[16 * 32];          // 32x32  -> 4KB
    __shared__ float2 sW2p[16 * 16];          // 32x16p -> 2KB
    __shared__ float sB[48];
    __shared__ __align__(16) float sTile[8][16 * 32];   // 16KB

    for (int i = threadIdx.x; i < 16 * 32; i += 256) {
        const int p = i >> 5, n = i & 31;
        float2 w;
        w.x = W1[(2 * p) * 32 + n];
        w.y = W1[(2 * p + 1) * 32 + n];
        sW1p[i] = w;
    }
    for (int i = threadIdx.x; i < 16 * 16; i += 256) {
        const int p = i >> 4, n = i & 15;
        float2 w;
        w.x = (n < 12) ? W2[(2 * p) * 12 + n] : 0.f;
        w.y = (n < 12) ? W2[(2 * p + 1) * 12 + n] : 0.f;
        sW2p[i] = w;
    }
    if (threadIdx.x < 48) {
        int i = threadIdx.x;
        sB[i] = (i < 32) ? b1[i] : (((i - 32) < 12) ? b2[i - 32] : 0.f);
    }
    __syncthreads();

    const int lane = threadIdx.x & 31;
    const int wave = threadIdx.x >> 5;
    const int col  = lane & 15;
    const int kh   = (lane >> 4) << 1;
    const int khp  = lane >> 4;
    const int rb   = (lane >> 4) << 3;
    const int mt   = blockIdx.x * 8 + wave;
    const int mtiles = (M + 15) >> 4;
    if (mt >= mtiles) return;

    int m = mt * 16 + col;
    if (m >= M) m = M - 1;
    float* tp = &sTile[wave][0];

    v8f acc[2] = {zero8(), zero8()};
#pragma unroll
    for (int kt = 0; kt < 8; ++kt) {
        const int k = kt * 4 + kh;
        const v2f a = ld2g(h + (size_t)m * 32 + k);
        const int pr = kt * 2 + khp;
#pragma unroll
        for (int nt = 0; nt < 2; ++nt)
            acc[nt] = wmma4(
                a, ld2s((const float*)&sW1p[pr * 32 + nt * 16 + col]),
                acc[nt]);
    }
#pragma unroll
    for (int nt = 0; nt < 2; ++nt) {
        const float bb = sB[nt * 16 + col];
#pragma unroll
        for (int v = 0; v < 8; ++v)
            tp[(rb + v) * 32 + nt * 16 + col] = fmaxf(acc[nt][v] + bb, 0.f);
    }
    wave_lds_fence();

    v8f acc2 = zero8();
#pragma unroll
    for (int kt = 0; kt < 8; ++kt) {
        const int k = kt * 4 + kh;
        const v2f a = ld2s(tp + col * 32 + k);
        const int pr = kt * 2 + khp;
        acc2 = wmma4(a, ld2s((const float*)&sW2p[pr * 16 + col]), acc2);
    }
    const float bb = sB[32 + col];
#pragma unroll
    for (int v = 0; v < 8; ++v) {
        const int r = mt * 16 + rb + v;
        if (r < M && col < 12)
            out[(size_t)r * 12 + col] = fmaxf(acc2[v] + bb, 0.f);
    }
}

// ---------------------------------------------------------------------------
// Host-side orchestration
// ---------------------------------------------------------------------------
extern "C" void kernel_launch(void* const* d_in, const int* in_sizes, int n_in,
                              void* d_out, int out_size, void* d_ws,
                              size_t ws_size, hipStream_t stream) {
    const float* x  = (const float*)d_in[0];
    const long long* ei = (const long long*)d_in[1];
    const float* ew = (const float*)d_in[2];
    const float *W1 = (const float*)d_in[3],  *b1 = (const float*)d_in[4];
    const float *W2 = (const float*)d_in[5],  *b2 = (const float*)d_in[6];
    const float *W3 = (const float*)d_in[7],  *b3 = (const float*)d_in[8];
    const float *W4 = (const float*)d_in[9],  *b4 = (const float*)d_in[10];
    const float *W5 = (const float*)d_in[11], *b5 = (const float*)d_in[12];
    const float *nhW1 = (const float*)d_in[13], *nhb1 = (const float*)d_in[14];
    const float *nhW2 = (const float*)d_in[15], *nhb2 = (const float*)d_in[16];
    const float *ehW1 = (const float*)d_in[17], *ehb1 = (const float*)d_in[18];
    const float *ehW2 = (const float*)d_in[19], *ehb2 = (const float*)d_in[20];
    const float *ehW3 = (const float*)d_in[21], *ehb3 = (const float*)d_in[22];
    const float *ehW4 = (const float*)d_in[23], *ehb4 = (const float*)d_in[24];

    const int N = in_sizes[0] / 128;
    const int E = in_sizes[2];
    const long long* src = ei;
    const long long* dstv = ei + E;

    // ws layout (floats): norm[E] | dinv[N] | bufA|bufB|bufC [N*128] | packed W
    float* wsf  = (float*)d_ws;
    float* norm = wsf;
    float* dinv = wsf + (size_t)E;
    size_t off  = (((size_t)E + (size_t)N) + 3) & ~(size_t)3;
    float* bufA = wsf + off;
    float* bufB = bufA + (size_t)N * 128;
    float* bufC = bufB + (size_t)N * 128;
    float* wp1  = bufC + (size_t)N * 128;     // 128x128 = 16384
    float* wp2  = wp1 + 16384;                // 128x64  =  8192
    float* wp3  = wp2 + 8192;                 // 64x64   =  4096
    float* wp4  = wp3 + 4096;                 // 64x32   =  2048
    float* wp5  = wp4 + 2048;                 // 32x32   =  1024

    float* nodeOut = (float*)d_out;
    float* edgeOut = nodeOut + (size_t)N * 12;

    const int T = 256;
    auto cdiv = [](long a, long b) { return (int)((a + b - 1) / b); };

    // ---- pack GCN weights into pair-interleaved layout ----
    k_pack_pairs<<<cdiv(64 * 128, T), T, 0, stream>>>(W1, wp1, 128, 128);
    k_pack_pairs<<<cdiv(64 * 64, T), T, 0, stream>>>(W2, wp2, 128, 64);
    k_pack_pairs<<<cdiv(32 * 64, T), T, 0, stream>>>(W3, wp3, 64, 64);
    k_pack_pairs<<<cdiv(32 * 32, T), T, 0, stream>>>(W4, wp4, 64, 32);
    k_pack_pairs<<<cdiv(16 * 32, T), T, 0, stream>>>(W5, wp5, 32, 32);

    // ---- gcn_norm ----
    k_deg_init<<<cdiv(N, T), T, 0, stream>>>(dinv, N);
    k_deg_acc<<<cdiv(E, T), T, 0, stream>>>(dstv, ew, dinv, E);
    k_rsqrt_inplace<<<cdiv(N, T), T, 0, stream>>>(dinv, N);
    k_norm<<<cdiv(E, T), T, 0, stream>>>(src, dstv, ew, dinv, norm, E);

    const int mtiles = (N + 15) >> 4;
    const int gemmBlocks = (mtiles + 7) / 8;

    auto agg_layer = [&](const float* t, float* agg, const float* bias,
                         int dout, int qlog) {
        const int nq = (int)((long)N * dout / 4);
        const int eq = (int)((long)E * dout / 4);
        k_selfinit<<<cdiv(nq, T), T, 0, stream>>>(t, dinv, agg, nq, qlog);
        k_scatter<<<cdiv(eq, T), T, 0, stream>>>(t, src, dstv, norm, agg, eq,
                                                 qlog, dout);
        k_bias_relu<<<cdiv(nq, T), T, 0, stream>>>(agg, bias, nq, qlog);
    };

    // L1: 128 -> 128    h_in = x           -> h1 = bufB
    k_gemm<32, 8><<<gemmBlocks, T, 0, stream>>>(x, wp1, bufA, N);
    agg_layer(bufA, bufB, b1, 128, 5);
    // L2: 128 -> 64     h1 = bufB          -> h2 = bufC
    k_gemm<32, 4><<<gemmBlocks, T, 0, stream>>>(bufB, wp2, bufA, N);
    agg_layer(bufA, bufC, b2, 64, 4);
    // L3: 64 -> 64      h2 = bufC          -> h3 = bufB
    k_gemm<16, 4><<<gemmBlocks, T, 0, stream>>>(bufC, wp3, bufA, N);
    agg_layer(bufA, bufB, b3, 64, 4);
    // L4: 64 -> 32      h3 = bufB          -> h4 = bufC
    k_gemm<16, 2><<<gemmBlocks, T, 0, stream>>>(bufB, wp4, bufA, N);
    agg_layer(bufA, bufC, b4, 32, 3);
    // L5: 32 -> 32      h4 = bufC          -> h5 = bufB
    k_gemm<8, 2><<<gemmBlocks, T, 0, stream>>>(bufC, wp5, bufA, N);
    agg_layer(bufA, bufB, b5, 32, 3);

    // heads (h = bufB, [N,32])
    k_node_head<<<gemmBlocks, T, 0, stream>>>(bufB, nhW1, nhb1, nhW2, nhb2,
                                              nodeOut, N);
    const int etiles = (E + 15) >> 4;
    k_edge_head<<<(etiles + 7) / 8, T, 0, stream>>>(
        bufB, src, dstv, ehW1, ehb1, ehW2, ehb2, ehW3, ehb3, ehW4, ehb4,
        edgeOut, E);
}